// Filter_43654047597028
// MI455X (gfx1250) — compile-verified
//
#include <hip/hip_runtime.h>

// MI455X / gfx1250: wave32, WMMA 16x16x4 f32 for the edge MLP,
// NT-streaming VALU phase for the per-edge heterogeneous matvec.

typedef __attribute__((ext_vector_type(2))) float v2f;
typedef __attribute__((ext_vector_type(4))) float v4f;
typedef __attribute__((ext_vector_type(8))) float v8f;

#define WAVES_PER_BLOCK 8
#define HPAD 36   // padded LDS row stride (floats): 144B = 16B-aligned, bank-spreading

__device__ __forceinline__ v8f wmma_f32(v2f a, v2f b, v8f c) {
    // D = A(16x4,f32) * B(4x16,f32) + C(16x16,f32)
    return __builtin_amdgcn_wmma_f32_16x16x4_f32(false, a, false, b,
                                                 (short)0, c, false, false);
}

// Fold the constant node latent through the first layer:
// b1p[n] = b1[n] + sum_{k<64} zei[k] * W1[k][n]
__global__ __launch_bounds__(32)
void bias_fold_kernel(const float* __restrict__ zei,
                      const float* __restrict__ W1,
                      const float* __restrict__ b1,
                      float* __restrict__ b1p)
{
    const int n = threadIdx.x;  // 0..31
    float acc = b1[n];
#pragma unroll 8
    for (int k = 0; k < 64; ++k)
        acc = fmaf(zei[k], W1[k * 32 + n], acc);
    b1p[n] = acc;
}

// Load per-lane B fragments for one 32x32 weight matrix (row-major W[k][n]).
// B layout (4x16 f32): lanes 0-15 -> N=0..15; VGPR0 = K rows {0,2}, VGPR1 = {1,3}.
__device__ __forceinline__ void load_bfrag(const float* __restrict__ W,
                                           int nn, int half,
                                           v2f B0[8], v2f B1[8])
{
#pragma unroll
    for (int j = 0; j < 8; ++j) {
        const int r = 4 * j + 2 * half;
        B0[j] = v2f{ W[(r    ) * 32 + nn     ], W[(r + 1) * 32 + nn     ] };
        B1[j] = v2f{ W[(r    ) * 32 + nn + 16], W[(r + 1) * 32 + nn + 16] };
    }
}

__global__ __launch_bounds__(256)
void edge_msg_kernel(const float* __restrict__ Mij,     // (N, 32)
                     const float* __restrict__ fw,      // (N, 32, 16)
                     const float* __restrict__ W1,      // (96, 32) -- rows 64..95 used
                     const float* __restrict__ W2,      // (32, 32)
                     const float* __restrict__ W3,      // (32, 32)
                     const float* __restrict__ b1p,     // (32,) folded bias
                     const float* __restrict__ b2,
                     const float* __restrict__ b3,
                     float* __restrict__ out,           // (N, 16)
                     int nEdges)
{
    __shared__ __align__(16) float sH[WAVES_PER_BLOCK][16][HPAD];

    const int lane = threadIdx.x & 31;
    const int wave = threadIdx.x >> 5;
    const int half = lane >> 4;   // which K/M half this lane serves
    const int nn   = lane & 15;   // row (A) / column (B,C,D) index

    long base = ((long)blockIdx.x * WAVES_PER_BLOCK + wave) * 16;
    if (base + 16 > (long)nEdges) base = (long)nEdges - 16;   // uniform per wave

    // ---- streaming-phase lane mapping + early prefetch of the f_weights tile ----
    const int   eoff  = lane >> 1;      // edge within tile (2 lanes per edge)
    const int   ohalf = lane & 1;       // which 8 outputs
    const float* fp = fw + (base + eoff) * 512 + ohalf * 8;
    __builtin_prefetch(fp,        0, 0);
    __builtin_prefetch(fp +  128, 0, 0);
    __builtin_prefetch(fp +  256, 0, 0);
    __builtin_prefetch(fp +  384, 0, 0);

    // ---- resident B fragments for all three layers (dead before streaming phase) ----
    v2f B1a[8], B1b[8], B2a[8], B2b[8], B3a[8], B3b[8];
    load_bfrag(W1 + 64 * 32, nn, half, B1a, B1b);   // effective 32x32 slice of W1
    load_bfrag(W2,           nn, half, B2a, B2b);
    load_bfrag(W3,           nn, half, B3a, B3b);
    const float bs1a = b1p[nn], bs1b = b1p[nn + 16];
    const float bs2a = b2 [nn], bs2b = b2 [nn + 16];
    const float bs3a = b3 [nn], bs3b = b3 [nn + 16];

    // ================= Layer 1: relu(Mij @ W1b + b1') =================
    // A layout (16x4 f32): lane holds row M=nn; VGPR0/1 = K {2*half, 2*half+1} per step.
    v8f c0 = {}, c1 = {};
    const float* arow = Mij + (base + nn) * 32 + 2 * half;
#pragma unroll
    for (int j = 0; j < 8; ++j) {
        v2f a = __builtin_nontemporal_load((const v2f*)(arow + 4 * j));
        c0 = wmma_f32(a, B1a[j], c0);
        c1 = wmma_f32(a, B1b[j], c1);
    }
#pragma unroll
    for (int v = 0; v < 8; ++v) {       // D layout: lane holds M = v + 8*half, N = nn(+16)
        sH[wave][v + 8 * half][nn     ] = fmaxf(c0[v] + bs1a, 0.0f);
        sH[wave][v + 8 * half][nn + 16] = fmaxf(c1[v] + bs1b, 0.0f);
    }
    __syncthreads();

    // ================= Layer 2: relu(h1 @ W2 + b2) =================
    c0 = v8f{}; c1 = v8f{};
#pragma unroll
    for (int j = 0; j < 8; ++j) {
        v2f a = *(const v2f*)&sH[wave][nn][4 * j + 2 * half];
        c0 = wmma_f32(a, B2a[j], c0);
        c1 = wmma_f32(a, B2b[j], c1);
    }
    __syncthreads();
#pragma unroll
    for (int v = 0; v < 8; ++v) {
        sH[wave][v + 8 * half][nn     ] = fmaxf(c0[v] + bs2a, 0.0f);
        sH[wave][v + 8 * half][nn + 16] = fmaxf(c1[v] + bs2b, 0.0f);
    }
    __syncthreads();

    // ================= Layer 3: h = h2 @ W3 + b3 (no relu) =================
    c0 = v8f{}; c1 = v8f{};
#pragma unroll
    for (int j = 0; j < 8; ++j) {
        v2f a = *(const v2f*)&sH[wave][nn][4 * j + 2 * half];
        c0 = wmma_f32(a, B3a[j], c0);
        c1 = wmma_f32(a, B3b[j], c1);
    }
    __syncthreads();
#pragma unroll
    for (int v = 0; v < 8; ++v) {
        sH[wave][v + 8 * half][nn     ] = c0[v] + bs3a;
        sH[wave][v + 8 * half][nn + 16] = c1[v] + bs3b;
    }
    __syncthreads();

    // ======= Streaming phase: message[e,o] = sum_d h[e,d] * f[e,d,o] =======
    // 2 lanes per edge; each lane owns 8 outputs, walks all 32 d's.
    float acc[8];
#pragma unroll
    for (int o = 0; o < 8; ++o) acc[o] = 0.0f;

#pragma unroll
    for (int q = 0; q < 8; ++q) {
        v4f hq = *(const v4f*)&sH[wave][eoff][4 * q];   // h[e, 4q..4q+3]
#pragma unroll
        for (int dd = 0; dd < 4; ++dd) {
            const int d = 4 * q + dd;
            v4f f0 = __builtin_nontemporal_load((const v4f*)(fp + d * 16    ));
            v4f f1 = __builtin_nontemporal_load((const v4f*)(fp + d * 16 + 4));
            const float hd = hq[dd];
#pragma unroll
            for (int o = 0; o < 4; ++o) {
                acc[o    ] = fmaf(hd, f0[o], acc[o    ]);
                acc[o + 4] = fmaf(hd, f1[o], acc[o + 4]);
            }
        }
    }

    float* op = out + (base + eoff) * 16 + ohalf * 8;
    __builtin_nontemporal_store(v4f{acc[0], acc[1], acc[2], acc[3]}, (v4f*)op);
    __builtin_nontemporal_store(v4f{acc[4], acc[5], acc[6], acc[7]}, (v4f*)(op + 4));
}

extern "C" void kernel_launch(void* const* d_in, const int* in_sizes, int n_in,
                              void* d_out, int out_size, void* d_ws, size_t ws_size,
                              hipStream_t stream) {
    const float* zei = (const float*)d_in[0];
    const float* Mij = (const float*)d_in[1];
    const float* fw  = (const float*)d_in[2];
    const float* W1  = (const float*)d_in[3];
    const float* b1  = (const float*)d_in[4];
    const float* W2  = (const float*)d_in[5];
    const float* b2  = (const float*)d_in[6];
    const float* W3  = (const float*)d_in[7];
    const float* b3  = (const float*)d_in[8];
    float* out = (float*)d_out;
    float* b1p = (float*)d_ws;            // 32 floats of scratch

    const int n = in_sizes[1] / 32;       // Mij is (N, 32)

    bias_fold_kernel<<<1, 32, 0, stream>>>(zei, W1, b1, b1p);

    const int nTiles  = (n + 15) / 16;
    const int nBlocks = (nTiles + WAVES_PER_BLOCK - 1) / WAVES_PER_BLOCK;
    edge_msg_kernel<<<nBlocks, 256, 0, stream>>>(Mij, fw, W1, W2, W3,
                                                 b1p, b2, b3, out, n);
}